// LSSL_35751307771916
// MI455X (gfx1250) — compile-verified
//
#include <hip/hip_runtime.h>

typedef __attribute__((ext_vector_type(2))) float v2f;
typedef __attribute__((ext_vector_type(8))) float v8f;

#define NBATCH 16
#define SEQL   2048
#define HID    256
#define NST    64
#define NCHAN  2
#define HM     (HID * NCHAN)       // 512
#define BL     (NBATCH * SEQL)     // 32768
#define AEXTW  68                  // 64 A_d cols + B_d col + 3 zero pad (K multiple of 4)
#define PITCH  20                  // LDS row pitch in dwords (bank-conflict-free)
#define NROWS  72                  // 64 state rows + two 4-row u buffers (double-buffered)

static __device__ __forceinline__ v8f wmma4(v2f a, v2f b, v8f c) {
  // D = A(16x4,f32) * B(4x16,f32) + C(16x16,f32)  -> v_wmma_f32_16x16x4_f32
  return __builtin_amdgcn_wmma_f32_16x16x4_f32(false, a, false, b, (short)0, c,
                                               false, false);
}

// ---------------------------------------------------------------------------
// Kernel 1: bilinear discretization via forward substitution (A is lower-tri).
// One block per h, 64 threads = 64 RHS columns. Writes A_ext = [A_d | B_d | 0]
// row-major (h, n, 68) so the scan kernel can load A-operands directly.
// ---------------------------------------------------------------------------
__global__ __launch_bounds__(64) void lssl_discretize(
    const float* __restrict__ A, const float* __restrict__ Bv,
    const float* __restrict__ dt, float* __restrict__ Aext) {
  const int h = blockIdx.x;
  const int j = threadIdx.x;  // column 0..63
  const float dth = dt[h];

  __shared__ float Xs[NST][NST + 1];
  __shared__ float xb[NST];

  for (int n = 0; n < NST; ++n) {
    float s = ((n == j) ? 1.0f : 0.0f) + 0.5f * dth * A[n * NST + j];  // rhs
    float sb = (j == 0) ? dth * Bv[n] : 0.0f;
    for (int k = 0; k < n; ++k) {
      const float lnk = -0.5f * dth * A[n * NST + k];  // lhs off-diagonal
      s -= lnk * Xs[k][j];
      if (j == 0) sb -= lnk * xb[k];
    }
    const float diag = 1.0f - 0.5f * dth * A[n * NST + n];
    Xs[n][j] = s / diag;
    if (j == 0) xb[n] = sb / diag;
    __syncthreads();
  }

  for (int n = 0; n < NST; ++n)
    Aext[(h * NST + n) * AEXTW + j] = Xs[n][j];
  // j indexes the row for the tail columns
  Aext[(h * NST + j) * AEXTW + 64] = xb[j];
  Aext[(h * NST + j) * AEXTW + 65] = 0.0f;
  Aext[(h * NST + j) * AEXTW + 66] = 0.0f;
  Aext[(h * NST + j) * AEXTW + 67] = 0.0f;
}

// ---------------------------------------------------------------------------
// Kernel 2: sequential scan. One wave per h. State S = x^T (64 x 16) lives in
// LDS between steps; A_ext and padded C live in registers in A-operand layout.
//   S_t = A_ext @ [S_{t-1}; u_t^T; 0]   (4x17 = 68 f32 WMMAs)
//   y_t = Cpad  @ S_t                    (16 f32 WMMAs)
// u_{t+1} is staged global->LDS with GLOBAL_LOAD_ASYNC_TO_LDS_B32 into a
// double-buffered row pair; the step's 68 state WMMAs cover the async latency and
// s_wait_asynccnt 0 lands just before the B-operand reload.
// GELU fused; Y written K-major (hm, b*L + t) for the projection GEMM.
// ---------------------------------------------------------------------------
__global__ __launch_bounds__(32) void lssl_scan(
    const float* __restrict__ in, const float* __restrict__ Aext,
    const float* __restrict__ Cmat, const float* __restrict__ Dmat,
    float* __restrict__ Yg) {
  const int h    = blockIdx.x;
  const int lane = threadIdx.x;
  const int half = lane >> 4;   // 0: lanes 0-15, 1: lanes 16-31
  const int l16  = lane & 15;

  __shared__ float Xs[NROWS * PITCH];  // rows 0..71, 16 cols used, pitch 20

  // ---- A_ext in A-operand layout: tile i rows 16i..16i+15, chunk c cols 4c..4c+3
  v2f Areg[4][17];
#pragma unroll
  for (int i = 0; i < 4; ++i) {
#pragma unroll
    for (int c = 0; c < 17; ++c) {
      const int m  = i * 16 + l16;
      const int k0 = c * 4 + 2 * half;  // A layout: v0 = {K, K+2}, v1 = {K+1, K+3}
      Areg[i][c].x = Aext[(h * NST + m) * AEXTW + k0];
      Areg[i][c].y = Aext[(h * NST + m) * AEXTW + k0 + 1];
    }
  }
  // ---- C padded to 16 rows (rows >= NCHAN are zero)
  v2f Creg[16];
#pragma unroll
  for (int c = 0; c < 16; ++c) {
    const int k0 = c * 4 + 2 * half;
    float c0 = 0.0f, c1 = 0.0f;
    if (l16 < NCHAN) {
      c0 = Cmat[(h * NCHAN + l16) * NST + k0];
      c1 = Cmat[(h * NCHAN + l16) * NST + k0 + 1];
    }
    Creg[c].x = c0;
    Creg[c].y = c1;
  }
  const float d0 = Dmat[h * NCHAN + 0];
  const float d1 = Dmat[h * NCHAN + 1];

  // ---- zero all LDS rows (state + both u buffers), then async-stage u_0
  for (int r = half; r < NROWS; r += 2) Xs[r * PITCH + l16] = 0.0f;
  __syncthreads();
  if (half == 0) {
    const unsigned ldsoff =
        (unsigned)(uintptr_t)&Xs[64 * PITCH + l16];        // LDS aperture: [31:0] = LDS offset
    const float* gp = &in[(l16 * SEQL + 0) * HID + h];
    asm volatile("global_load_async_to_lds_b32 %0, %1, off"
                 :: "v"(ldsoff), "v"(gp) : "memory");
  }
  asm volatile("s_wait_asynccnt 0" ::: "memory");
  __syncthreads();

  // ---- preload B operands (chunk c = rows 4c..4c+3; chunk 16 = u buffer 0)
  v2f bop[17];
#pragma unroll
  for (int c = 0; c < 17; ++c) {
    const int r0 = 4 * c + 2 * half;
    bop[c].x = Xs[r0 * PITCH + l16];
    bop[c].y = Xs[(r0 + 1) * PITCH + l16];
  }

  const v8f zero8 = {};
#pragma unroll 1
  for (int t = 0; t < SEQL; ++t) {
    const float u_t = bop[16].x;  // u row: valid in lanes 0-15 (half==0)

    // kick off the async stage of u_{t+1} into the other buffer
    const int ubnext = 64 + 4 * ((t + 1) & 1);
    if (t + 1 < SEQL) {
      if (half == 0) {
        const unsigned ldsoff =
            (unsigned)(uintptr_t)&Xs[ubnext * PITCH + l16];
        const float* gp = &in[(l16 * SEQL + (t + 1)) * HID + h];
        asm volatile("global_load_async_to_lds_b32 %0, %1, off"
                     :: "v"(ldsoff), "v"(gp) : "memory");
      }
    }

    // state update: S_t = A_ext @ [S_{t-1}; u_t; 0]
    v8f st[4];
#pragma unroll
    for (int i = 0; i < 4; ++i) {
      v8f acc = zero8;
#pragma unroll
      for (int c = 0; c < 17; ++c) acc = wmma4(Areg[i][c], bop[c], acc);
      st[i] = acc;
    }
    // write S_t back to LDS (D layout: vgpr r -> rows 16i+r / 16i+r+8)
#pragma unroll
    for (int i = 0; i < 4; ++i)
#pragma unroll
      for (int r = 0; r < 8; ++r)
        Xs[(16 * i + r + 8 * half) * PITCH + l16] = st[i][r];

    asm volatile("s_wait_asynccnt 0" ::: "memory");
    __syncthreads();

    // reload B operands (x_t for y-WMMA now and for the next state step,
    // plus u_{t+1} from the freshly filled buffer)
#pragma unroll
    for (int c = 0; c < 16; ++c) {
      const int r0 = 4 * c + 2 * half;
      bop[c].x = Xs[r0 * PITCH + l16];
      bop[c].y = Xs[(r0 + 1) * PITCH + l16];
    }
    {
      const int r0 = ubnext + 2 * half;
      bop[16].x = Xs[r0 * PITCH + l16];
      bop[16].y = Xs[(r0 + 1) * PITCH + l16];
    }

    // y_t = Cpad @ S_t (chunks 0..15 only: exclude the u row)
    v8f ya = zero8;
#pragma unroll
    for (int c = 0; c < 16; ++c) ya = wmma4(Creg[c], bop[c], ya);

    float y0 = ya[0] + d0 * u_t;  // m=0, lanes 0-15 hold batch b=l16
    float y1 = ya[1] + d1 * u_t;  // m=1
    const float g0 = 0.5f * y0 * (1.0f + erff(y0 * 0.70710678118654752f));
    const float g1 = 0.5f * y1 * (1.0f + erff(y1 * 0.70710678118654752f));
    if (half == 0) {
      Yg[((h * NCHAN + 0) * NBATCH + l16) * SEQL + t] = g0;  // = Yg[k*BL + bl]
      Yg[((h * NCHAN + 1) * NBATCH + l16) * SEQL + t] = g1;
    }
  }
}

// ---------------------------------------------------------------------------
// Kernel 3: output projection  out(BL x 256) = Yg^T(BL x 512) @ W(512 x 256)
// Yg is stored K-major (512 x BL). One wave computes a 32x64 strip: two
// M-tiles share every B operand, halving redundant W/Yg L2 traffic per WMMA.
// ---------------------------------------------------------------------------
__global__ __launch_bounds__(32) void lssl_proj(
    const float* __restrict__ Yg, const float* __restrict__ W,
    float* __restrict__ out) {
  const int lane = threadIdx.x;
  const int half = lane >> 4;
  const int l16  = lane & 15;
  const int m0   = blockIdx.x * 32;   // M rows (b*L + t)
  const int n0   = blockIdx.y * 64;   // N block

  v8f acc[2][4] = {};
#pragma unroll 4
  for (int c = 0; c < HM / 4; ++c) {
    const int k0 = 4 * c + 2 * half;
    v2f a0, a1;
    a0.x = Yg[k0 * BL + m0 + l16];
    a0.y = Yg[(k0 + 1) * BL + m0 + l16];
    a1.x = Yg[k0 * BL + m0 + 16 + l16];
    a1.y = Yg[(k0 + 1) * BL + m0 + 16 + l16];
#pragma unroll
    for (int j = 0; j < 4; ++j) {
      const int n = n0 + j * 16 + l16;
      v2f b;
      b.x = W[k0 * HID + n];
      b.y = W[(k0 + 1) * HID + n];
      acc[0][j] = wmma4(a0, b, acc[0][j]);
      acc[1][j] = wmma4(a1, b, acc[1][j]);
    }
  }
#pragma unroll
  for (int p = 0; p < 2; ++p)
#pragma unroll
    for (int j = 0; j < 4; ++j)
#pragma unroll
      for (int r = 0; r < 8; ++r)
        out[(m0 + p * 16 + r + 8 * half) * HID + n0 + j * 16 + l16] =
            acc[p][j][r];
}

extern "C" void kernel_launch(void* const* d_in, const int* in_sizes, int n_in,
                              void* d_out, int out_size, void* d_ws,
                              size_t ws_size, hipStream_t stream) {
  const float* in   = (const float*)d_in[0];  // (16, 2048, 256)
  const float* A    = (const float*)d_in[1];  // (64, 64)
  const float* Bv   = (const float*)d_in[2];  // (64,)
  const float* dt   = (const float*)d_in[3];  // (256,)
  const float* Cmat = (const float*)d_in[4];  // (256, 2, 64)
  const float* Dmat = (const float*)d_in[5];  // (256, 2)
  const float* W    = (const float*)d_in[6];  // (512, 256)
  float* out        = (float*)d_out;          // (16, 2048, 256)

  float* Aext = (float*)d_ws;                       // 256*64*68 floats (~4.5 MB)
  float* Yg   = Aext + (size_t)HID * NST * AEXTW;   // 512*32768 floats (64 MB)

  lssl_discretize<<<HID, 64, 0, stream>>>(A, Bv, dt, Aext);
  lssl_scan<<<HID, 32, 0, stream>>>(in, Aext, Cmat, Dmat, Yg);
  dim3 grid(BL / 32, HID / 64);
  lssl_proj<<<grid, 32, 0, stream>>>(Yg, W, out);
}